// GCN_11046655885806
// MI455X (gfx1250) — compile-verified
//
#include <hip/hip_runtime.h>

typedef __attribute__((ext_vector_type(2))) float v2f;
typedef __attribute__((ext_vector_type(8))) float v8f;

#define NF 128          // feature width for all layers
#define NTILES (NF/16)  // 8 column tiles

// ---------------- bias broadcast init: dst[i*NF+f] = bias[f] ----------------
__global__ void bias_init_kernel(float* __restrict__ dst,
                                 const float* __restrict__ bias, int total) {
  int idx = blockIdx.x * blockDim.x + threadIdx.x;
  if (idx < total) dst[idx] = bias[idx & (NF - 1)];
}

// ---------------- in-place ReLU ----------------
__global__ void relu_kernel(float* __restrict__ x, int total) {
  int idx = blockIdx.x * blockDim.x + threadIdx.x;
  if (idx < total) x[idx] = fmaxf(x[idx], 0.0f);
}

// ---------------- dense GEMM: C[M x NF] = (relu?)(A[M x NF]) * W[NF x NF] ----
// One wave computes one 16x16 output tile via 32x V_WMMA_F32_16X16X4_F32.
// Wave id -> (mtile, ntile). All dims divide exactly; EXEC is all-ones.
template <bool RELU_IN>
__global__ void gemm_wmma_f32_kernel(const float* __restrict__ A,
                                     const float* __restrict__ W,
                                     float* __restrict__ C, int M) {
  const int lane = threadIdx.x & 31;
  const int wave = blockIdx.x * (blockDim.x >> 5) + (threadIdx.x >> 5);
  const int mtile = wave >> 3;       // / NTILES
  const int ntile = wave & 7;        // % NTILES
  if (mtile * 16 >= M) return;       // wave-uniform guard

  const int m16   = lane & 15;
  const int khalf = (lane >> 4) << 1;          // 0 for lanes 0-15, 2 for 16-31
  const int col   = ntile * 16 + m16;
  const float* __restrict__ arow = A + (long)(mtile * 16 + m16) * NF;

  v8f c = {};
  #pragma unroll
  for (int k0 = 0; k0 < NF; k0 += 4) {
    // A fragment (16x4 f32): lane -> row m16, K = k0 + khalf + {0,1}
    v2f a;
    a.x = arow[k0 + khalf];
    a.y = arow[k0 + khalf + 1];
    if (RELU_IN) { a.x = fmaxf(a.x, 0.0f); a.y = fmaxf(a.y, 0.0f); }
    // B fragment (4x16 f32): lane -> col, K = k0 + khalf + {0,1}
    v2f b;
    b.x = W[(long)(k0 + khalf) * NF + col];
    b.y = W[(long)(k0 + khalf + 1) * NF + col];
    c = __builtin_amdgcn_wmma_f32_16x16x4_f32(
        /*neg_a=*/false, a, /*neg_b=*/false, b,
        /*c_mod=*/(short)0, c, /*reuse_a=*/false, /*reuse_b=*/false);
  }

  // C/D layout: VGPR r holds M = r + 8*(lane>=16), N = lane%16
  const int rowBase = mtile * 16 + ((lane >> 4) << 3);
  #pragma unroll
  for (int r = 0; r < 8; ++r)
    C[(long)(rowBase + r) * NF + col] = c[r];
}

// ---------------- SpMM scatter: dst[rcv[e]] += ev[e] * sup[snd[e]] ----------
// One wave per edge; each lane owns 4 consecutive features (float4 gather,
// 4x global_atomic_add_f32 scatter). Edge scalars are wave-uniform loads.
__global__ void spmm_scatter_kernel(const float* __restrict__ sup,
                                    const int* __restrict__ snd,
                                    const int* __restrict__ rcv,
                                    const float* __restrict__ ev,
                                    float* __restrict__ dst, int nedges) {
  long tid = (long)blockIdx.x * blockDim.x + threadIdx.x;
  const long stride = (long)gridDim.x * blockDim.x;
  const long total = (long)nedges * 32;
  for (; tid < total; tid += stride) {
    const long e = tid >> 5;
    const int lane = (int)(tid & 31);
    const int s = __builtin_amdgcn_readfirstlane(snd[e]);
    const int r = __builtin_amdgcn_readfirstlane(rcv[e]);
    const float w = ev[e];
    const float4 v = *(const float4*)(sup + (long)s * NF + lane * 4);
    float* o = dst + (long)r * NF + lane * 4;
    unsafeAtomicAdd(o + 0, v.x * w);
    unsafeAtomicAdd(o + 1, v.y * w);
    unsafeAtomicAdd(o + 2, v.z * w);
    unsafeAtomicAdd(o + 3, v.w * w);
  }
}

extern "C" void kernel_launch(void* const* d_in, const int* in_sizes, int n_in,
                              void* d_out, int out_size, void* d_ws, size_t ws_size,
                              hipStream_t stream) {
  const float* x    = (const float*)d_in[0];
  const int*   snd  = (const int*)  d_in[1];
  const int*   rcv  = (const int*)  d_in[2];
  const float* ev   = (const float*)d_in[3];
  const float* W1   = (const float*)d_in[4];
  const float* b1   = (const float*)d_in[5];
  const float* W2   = (const float*)d_in[6];
  const float* b2   = (const float*)d_in[7];
  float* out = (float*)d_out;

  const int N = in_sizes[0] / NF;        // 100000
  const int E = in_sizes[1];             // 3200000
  const int total = N * NF;              // 12.8M

  float* sup = (float*)d_ws;                         // [N, NF] support buffer
  float* h1  = (float*)d_ws + (size_t)N * NF;        // [N, NF] hidden buffer

  const int eltBlocks  = (total + 255) / 256;
  const int gemmBlocks = (N / 16) * NTILES / 4;      // 4 waves (128 thr) / block
  const int scatBlocks = 32768;                      // grid-stride over E*32

  // Layer 1: h1 = relu(spmm(x@W1) + b1)   (relu deferred into layer-2 A load)
  bias_init_kernel<<<eltBlocks, 256, 0, stream>>>(h1, b1, total);
  gemm_wmma_f32_kernel<false><<<gemmBlocks, 128, 0, stream>>>(x, W1, sup, N);
  spmm_scatter_kernel<<<scatBlocks, 256, 0, stream>>>(sup, snd, rcv, ev, h1, E);

  // Layer 2: out = relu(spmm(relu(h1)@W2) + b2)
  gemm_wmma_f32_kernel<true><<<gemmBlocks, 128, 0, stream>>>(h1, W2, sup, N);
  bias_init_kernel<<<eltBlocks, 256, 0, stream>>>(out, b2, total);
  spmm_scatter_kernel<<<scatBlocks, 256, 0, stream>>>(sup, snd, rcv, ev, out, E);
  relu_kernel<<<eltBlocks, 256, 0, stream>>>(out, total);
}